// BC_evidence_X_Estimation_bf_56358560858221
// MI455X (gfx1250) — compile-verified
//
#include <hip/hip_runtime.h>
#include <hip/hip_bf16.h>
#include <stdint.h>

// Problem constants (match reference)
#define TT 256
#define NN 100000
#define EE 8192
#define MU_PLUS  0.1f
#define MU_MINUS 0.05f
#define RHO      16.0f
#define CLIP_LO  1e-5f
#define CLIP_HI  (1.0f - 1e-5f)

// Persistent-kernel shape: 64 workgroups x 512 threads (16 wave32 each).
// 64 resident workgroups is far below MI455X WGP count, so the grid barrier
// cannot deadlock; 32768 threads give at most 1 node-vec4 iteration per phase.
#define GRID_BLOCKS   64
#define BLOCK_THREADS 512
#define NTH (GRID_BLOCKS * BLOCK_THREADS)

__device__ __forceinline__ float sigmoidf(float x) {
  return 1.0f / (1.0f + __expf(-x));
}
__device__ __forceinline__ float clipf(float x) {
  return fminf(fmaxf(x, CLIP_LO), CLIP_HI);
}

// Sense-reversing grid barrier. Thread 0 of each block arrives via an
// agent-scope atomic; the acquire on the spin load invalidates this WGP's
// vector cache so all threads observe other blocks' stores after the barrier.
__device__ __forceinline__ void grid_barrier(unsigned* cnt, unsigned* gen) {
  __threadfence();                 // flush this thread's stores to device scope
  __syncthreads();
  if (threadIdx.x == 0) {
    unsigned g = __hip_atomic_load(gen, __ATOMIC_RELAXED, __HIP_MEMORY_SCOPE_AGENT);
    unsigned prev = __hip_atomic_fetch_add(cnt, 1u, __ATOMIC_ACQ_REL, __HIP_MEMORY_SCOPE_AGENT);
    if (prev == GRID_BLOCKS - 1u) {
      __hip_atomic_store(cnt, 0u, __ATOMIC_RELAXED, __HIP_MEMORY_SCOPE_AGENT);
      __hip_atomic_store(gen, g + 1u, __ATOMIC_RELEASE, __HIP_MEMORY_SCOPE_AGENT);
    } else {
      while (__hip_atomic_load(gen, __ATOMIC_ACQUIRE, __HIP_MEMORY_SCOPE_AGENT) == g) {
        __builtin_amdgcn_s_sleep(2);   // low-power spin
      }
    }
  }
  __syncthreads();
}

__global__ void bc_init_barrier(unsigned* bar) {
  bar[0] = 0u;   // arrival counter (ws is poisoned 0xAA once; re-init every call)
  bar[1] = 0u;   // generation
}

__global__ void __launch_bounds__(BLOCK_THREADS, 1)
bc_persistent(const float* __restrict__ logit,
              const float* __restrict__ theta,
              const int*   __restrict__ ug,
              const int*   __restrict__ vg,
              const float* __restrict__ spg,
              const float* __restrict__ smg,
              float* __restrict__ Xout,   // [TT, NN]
              float* __restrict__ kp,     // [TT*EE]
              float* __restrict__ km,     // [TT*EE]
              float* __restrict__ wd,     // scratch [EE]
              unsigned* __restrict__ bar) {
  const int tid = (int)(blockIdx.x * BLOCK_THREADS + threadIdx.x);

  // CDNA5 async-copy staging buffers for this block's edge slice (8 KB LDS).
  __shared__ int   sh_u[BLOCK_THREADS];
  __shared__ int   sh_v[BLOCK_THREADS];
  __shared__ float sh_sp[BLOCK_THREADS];
  __shared__ float sh_sm[BLOCK_THREADS];

  const float ep = sigmoidf(theta[0]) * 0.5f;          // epsilon_plus
  const float em = 0.5f + sigmoidf(theta[1]) * 0.5f;   // epsilon_minus

  // ---- init: X[0] = sigmoid(logit_X0), UNCLIPPED (matches reference) ----
  {
    const float4* lg4 = (const float4*)logit;
    float4* x04 = (float4*)Xout;
    for (int i = tid; i < NN / 4; i += NTH) {
      float4 l = lg4[i];
      float4 s;
      s.x = sigmoidf(l.x); s.y = sigmoidf(l.y);
      s.z = sigmoidf(l.z); s.w = sigmoidf(l.w);
      x04[i] = s;
    }
  }
  grid_barrier(bar, bar + 1);

  // ---- main serial scan: 2 grid barriers per step ----
  for (int t = 1; t < TT; ++t) {
    float* rowp = Xout + (size_t)(t - 1) * NN;   // state at t-1 (raw if t>1)
    float* rowc = Xout + (size_t)t * NN;
    const int eb = (t - 1) * EE;
    const bool doclip = (t > 1);   // step 1 reads raw X0, per reference

    // Phase A: (a) finalize row t-1 with clip + copy forward to row t,
    //          (b) gather edges of step t-1 -> wd[] AND kappa row t-1.
    {
      float4* rp4 = (float4*)rowp;
      float4* rc4 = (float4*)rowc;
      for (int i = tid; i < NN / 4; i += NTH) {
        float4 x = rp4[i];
        if (doclip) {
          x.x = clipf(x.x); x.y = clipf(x.y);
          x.z = clipf(x.z); x.w = clipf(x.w);
          rp4[i] = x;                      // row t-1 output becomes clipped
        }
        rc4[i] = x;                        // working copy for row t
      }
      for (int e = tid; e < EE; e += NTH) {
        // Stage this edge's tuple through LDS with gfx1250 async loads
        // (tracked by ASYNCcnt, waited with s_wait_asynccnt).
        unsigned lu  = (unsigned)(uintptr_t)&sh_u[threadIdx.x];
        unsigned lv  = (unsigned)(uintptr_t)&sh_v[threadIdx.x];
        unsigned lsp = (unsigned)(uintptr_t)&sh_sp[threadIdx.x];
        unsigned lsm = (unsigned)(uintptr_t)&sh_sm[threadIdx.x];
        uint64_t gu  = (uint64_t)(uintptr_t)(ug  + eb + e);
        uint64_t gv  = (uint64_t)(uintptr_t)(vg  + eb + e);
        uint64_t gsp = (uint64_t)(uintptr_t)(spg + eb + e);
        uint64_t gsm = (uint64_t)(uintptr_t)(smg + eb + e);
        asm volatile("global_load_async_to_lds_b32 %0, %1, off" :: "v"(lu),  "v"(gu)  : "memory");
        asm volatile("global_load_async_to_lds_b32 %0, %1, off" :: "v"(lv),  "v"(gv)  : "memory");
        asm volatile("global_load_async_to_lds_b32 %0, %1, off" :: "v"(lsp), "v"(gsp) : "memory");
        asm volatile("global_load_async_to_lds_b32 %0, %1, off" :: "v"(lsm), "v"(gsm) : "memory");
#if defined(__has_builtin) && __has_builtin(__builtin_amdgcn_s_wait_asynccnt)
        __builtin_amdgcn_s_wait_asynccnt(0);
#else
        asm volatile("s_wait_asynccnt 0" ::: "memory");
#endif
        int uu = sh_u[threadIdx.x];
        int vv = sh_v[threadIdx.x];
        float xu = rowp[uu];
        float xv = rowp[vv];
        if (doclip) { xu = clipf(xu); xv = clipf(xv); }   // race-proof: clip idempotent
        float w = MU_PLUS * sh_sp[threadIdx.x] - MU_MINUS * sh_sm[threadIdx.x];
        wd[e] = w * (xv - xu);
        float ad = fabsf(xu - xv);
        kp[eb + e] = sigmoidf( RHO * (ep - ad));
        km[eb + e] = sigmoidf(-RHO * (em - ad));
        // Warm next step's edge tuples into GL2 (global_prefetch_b8).
        __builtin_prefetch(ug  + eb + EE + e, 0, 1);
        __builtin_prefetch(vg  + eb + EE + e, 0, 1);
        __builtin_prefetch(spg + eb + EE + e, 0, 1);
        __builtin_prefetch(smg + eb + EE + e, 0, 1);
      }
    }
    grid_barrier(bar, bar + 1);

    // Phase B: scatter the precomputed deltas into row t (L2 float atomics).
    for (int e = tid; e < EE; e += NTH) {
      int uu = ug[eb + e];
      int vv = vg[eb + e];
      float d = wd[e];
      atomicAdd(&rowc[uu],  d);
      atomicAdd(&rowc[vv], -d);
    }
    grid_barrier(bar, bar + 1);
  }

  // ---- epilogue: clip final row in-place + kappa for row T-1 ----
  {
    float* rowl = Xout + (size_t)(TT - 1) * NN;
    float4* rl4 = (float4*)rowl;
    for (int i = tid; i < NN / 4; i += NTH) {
      float4 x = rl4[i];
      x.x = clipf(x.x); x.y = clipf(x.y);
      x.z = clipf(x.z); x.w = clipf(x.w);
      rl4[i] = x;
    }
    const int eb = (TT - 1) * EE;
    for (int e = tid; e < EE; e += NTH) {
      float xu = clipf(rowl[ug[eb + e]]);   // clip-on-read: immune to the
      float xv = clipf(rowl[vg[eb + e]]);   // concurrent in-place clip above
      float ad = fabsf(xu - xv);
      kp[eb + e] = sigmoidf( RHO * (ep - ad));
      km[eb + e] = sigmoidf(-RHO * (em - ad));
    }
  }
}

extern "C" void kernel_launch(void* const* d_in, const int* in_sizes, int n_in,
                              void* d_out, int out_size, void* d_ws, size_t ws_size,
                              hipStream_t stream) {
  (void)in_sizes; (void)n_in; (void)out_size; (void)ws_size;
  const float* logit = (const float*)d_in[0];
  const float* theta = (const float*)d_in[1];
  const int*   u     = (const int*)  d_in[2];
  const int*   v     = (const int*)  d_in[3];
  const float* sp    = (const float*)d_in[4];
  const float* sm    = (const float*)d_in[5];

  float* Xout = (float*)d_out;                    // [TT*NN]
  float* kp   = Xout + (size_t)TT * NN;           // [TT*EE]
  float* km   = kp   + (size_t)TT * EE;           // [TT*EE]

  unsigned* bar = (unsigned*)d_ws;                // 2 x u32 barrier state
  float*    wd  = (float*)((char*)d_ws + 256);    // EE floats edge-delta scratch

  bc_init_barrier<<<1, 1, 0, stream>>>(bar);
  bc_persistent<<<GRID_BLOCKS, BLOCK_THREADS, 0, stream>>>(
      logit, theta, u, v, sp, sm, Xout, kp, km, wd, bar);
}